// PersistentHomologyLayer_37907381354922
// MI455X (gfx1250) — compile-verified
//
#include <hip/hip_runtime.h>
#include <hip/hip_bf16.h>
#include <math.h>

// CDNA5 / gfx1250, wave32. One workgroup per batch; Prim's MST is a serial
// chain so the goal is per-step latency, with the einsum row offloaded to
// V_WMMA_F32_16X16X4_F32 on the matrix pipe.

typedef __attribute__((ext_vector_type(2))) float v2f;
typedef __attribute__((ext_vector_type(8))) float v8f;

#define NPTS    2048
#define NBATCH  16
#define NFILT   50
#define MAXP    100
#define THREADS 256
#define GROUPS  16      // 16x16x4 WMMA tiles per wave
#define CHUNK   256     // points per wave (8 waves * 256 = 2048)

__global__ __launch_bounds__(THREADS)
void prim_ph_fused(const float* __restrict__ x,
                   const float* __restrict__ filt,
                   float* __restrict__ out)
{
    __shared__ float xx[NPTS], xy[NPTS], xz[NPTS];  // SoA coords
    __shared__ float sq[NPTS];                      // |x|^2
    __shared__ float deaths[NPTS];                  // N-1 = 2047 used
    __shared__ float wredv[2][8];                   // double-buffered cross-wave argmin
    __shared__ int   wredi[2][8];

    const int b    = blockIdx.x;
    const int t    = threadIdx.x;
    const int wave = t >> 5;
    const int lane = t & 31;
    const int col  = lane & 15;          // WMMA column owned by this lane
    const int wbase = wave * CHUNK;

    const float* xb = x + (size_t)b * NPTS * 3;
    __builtin_prefetch(xb, 0, 1);        // global_prefetch_b8 (gfx1250 path)

    // Stage x[b] into LDS, AoS -> SoA.
    for (int f = t; f < NPTS * 3; f += THREADS) {
        float v = xb[f];
        int n = f / 3, c = f - 3 * n;
        if (c == 0) xx[n] = v; else if (c == 1) xy[n] = v; else xz[n] = v;
    }
    __syncthreads();
    for (int n = t; n < NPTS; n += THREADS)
        sq[n] = xx[n] * xx[n] + xy[n] * xy[n] + xz[n] * xz[n];
    __syncthreads();

    // Persistent WMMA B operands: group g holds the 4x16 coord matrix of
    // points [wbase + g*16, wbase + g*16 + 16). ISA B layout (4 rows over
    // 2 VGPRs): VGPR0 = {K0 | K2}, VGPR1 = {K1 | K3=0} across half-waves.
    v2f   vB[GROUPS];
    float pmind[GROUPS];                 // Prim min_dist for owned points
    float psq[GROUPS];
    unsigned int treemask = 0;           // in-tree bit per owned group slot
    #pragma unroll
    for (int g = 0; g < GROUPS; ++g) {
        int p = wbase + g * 16 + col;
        vB[g].x = (lane < 16) ? xx[p] : xz[p];
        vB[g].y = (lane < 16) ? xy[p] : 0.0f;
        psq[g]  = sq[p];
        pmind[g] = __builtin_inff();
    }

    // Relaxation: dots of x[j] vs all owned points via WMMA, then the
    // sqrt(max(sq_j + sq_k - 2*dot, 0)) tail on the VALU. Raw v_sqrt_f32
    // (no denorm fixup): inputs are O(1) squared distances, never subnormal.
    auto relax = [&](int j) {
        float jx = xx[j], jy = xy[j], jz = xz[j], jsq = sq[j];
        v2f A;                           // all 16 rows = x[j] (K3 pad = 0)
        A.x = (lane < 16) ? jx : jz;
        A.y = (lane < 16) ? jy : 0.0f;
        #pragma unroll
        for (int g = 0; g < GROUPS; ++g) {
            v8f c = {};
            c = __builtin_amdgcn_wmma_f32_16x16x4_f32(
                    false, A, false, vB[g], (short)0, c, false, false);
            float dot = c[0];            // D[0, col] = x[j] . x[p]
            float d2  = fmaf(-2.0f, dot, jsq + psq[g]);
            float d   = __builtin_amdgcn_sqrtf(fmaxf(d2, 0.0f));
            pmind[g]  = fminf(pmind[g], d);
        }
    };

    relax(0);                            // min_dist = D[0]
    if (wave == 0 && col == 0) treemask |= 1u;   // point 0 in tree

    // ---- Prim main loop: 2047 serial extract-min / relax steps ----
    for (int s = 0; s < NPTS - 1; ++s) {
        // lane-local masked argmin (ascending g keeps lowest index on ties)
        float bv = __builtin_inff();
        int   bi = 0x7fffffff;
        #pragma unroll
        for (int g = 0; g < GROUPS; ++g) {
            float v = ((treemask >> g) & 1u) ? __builtin_inff() : pmind[g];
            int idx = wbase + g * 16 + col;
            if (v < bv || (v == bv && idx < bi)) { bv = v; bi = idx; }
        }
        // wave32 butterfly argmin (tie -> lowest index, matches jnp.argmin)
        #pragma unroll
        for (int m = 16; m >= 1; m >>= 1) {
            float ov = __shfl_xor(bv, m, 32);
            int   oi = __shfl_xor(bi, m, 32);
            if (ov < bv || (ov == bv && oi < bi)) { bv = ov; bi = oi; }
        }
        const int par = s & 1;
        if (lane == 0) { wredv[par][wave] = bv; wredi[par][wave] = bi; }
        __syncthreads();
        // every thread reduces the 8 wave results (avoids a second barrier)
        float wv = wredv[par][0];
        int   wj = wredi[par][0];
        #pragma unroll
        for (int u = 1; u < 8; ++u) {
            float ov = wredv[par][u]; int oi = wredi[par][u];
            if (ov < wv || (ov == wv && oi < wj)) { wv = ov; wj = oi; }
        }
        if (t == 0) deaths[s] = wv;      // MST edge weight = H0 death
        if (wj >= wbase && wj < wbase + CHUNK && (wj & 15) == col)
            treemask |= 1u << ((wj - wbase) >> 4);
        relax(wj);
    }
    __syncthreads();

    // ---- Epilogue: top-100 by stable rank + diagram padding ----
    float* diag = out + (size_t)b * 3 * MAXP * 2;
    for (int i = t; i < NPTS - 1; i += THREADS) {
        float di = deaths[i];
        int rank = 0;
        for (int j = 0; j < NPTS - 1; ++j) {     // LDS broadcast reads
            float dj = deaths[j];
            rank += (dj > di) || (dj == di && j < i);
        }
        if (rank < MAXP) {                        // descending, ties stable
            diag[rank * 2 + 0] = 0.0f;            // birth
            diag[rank * 2 + 1] = di;              // death
        }
    }
    for (int k = t; k < 2 * MAXP * 2; k += THREADS)   // H1/H2 zero pads
        diag[MAXP * 2 + k] = 0.0f;

    // ---- Betti-0 curve: N - #(deaths <= eps) ----
    float* bc = out + (size_t)NBATCH * 3 * MAXP * 2 + (size_t)b * 3 * NFILT;
    if (t < NFILT) {
        float f = filt[t];
        int cnt = 0;
        for (int j = 0; j < NPTS - 1; ++j) cnt += (deaths[j] <= f);
        bc[t] = (float)(NPTS - cnt);
    }
    for (int k = t; k < 2 * NFILT; k += THREADS)      // H1/H2 zero curves
        bc[NFILT + k] = 0.0f;
}

extern "C" void kernel_launch(void* const* d_in, const int* in_sizes, int n_in,
                              void* d_out, int out_size, void* d_ws, size_t ws_size,
                              hipStream_t stream) {
    const float* x    = (const float*)d_in[0];   // [16, 2048, 3] f32
    const float* filt = (const float*)d_in[1];   // [50] f32
    float* out = (float*)d_out;                  // 9600 diagram + 2400 betti f32
    (void)in_sizes; (void)n_in; (void)out_size; (void)d_ws; (void)ws_size;
    prim_ph_fused<<<NBATCH, THREADS, 0, stream>>>(x, filt, out);
}